// IterativeLayer_42485816492474
// MI455X (gfx1250) — compile-verified
//
#include <hip/hip_runtime.h>

#define NUMPIX   256
#define NUMBIN   367
#define NUMTHETA 360
#define NS       4
#define NANG     90            // angles per subset
#define CDET     183.0f        // (NUMBIN-1)/2
#define CPIX     127.5f        // (NUMPIX-1)/2
#define DEN_EPS  1e-6f
#define EPS64    2.2204460492503131e-16f
#define STILES   23            // ceil(367/16)
#define TCHUNKS  92            // ceil(367/4)
#define FP_N     (NANG*NUMBIN) // 33030
#define FP_PAD   33032         // FP_N rounded up to a 16B multiple (floats)
#define IMG_N    (NUMPIX*NUMPIX)

typedef __attribute__((ext_vector_type(2))) float v2f;
typedef __attribute__((ext_vector_type(8))) float v8f;

enum { F_MINV = 0, F_DIFFS = 1, F_RES = 2 };
enum { B_DINV = 0, B_SART = 1 };
enum { U_COPYPREV = 0, U_FMAXEPS = 1, U_SELECT = 2, U_OUT = 3 };

// Stage nbytes (multiple of 16) from global to LDS via the CDNA5 async
// memory->LDS path (GLOBAL_LOAD_ASYNC_TO_LDS_B128, GVS addressing, ASYNCcnt).
__device__ __forceinline__ void async_stage(const float* __restrict__ src,
                                            float* lds_dst, int nbytes) {
  unsigned lds_base = (unsigned)(uintptr_t)lds_dst;       // addr[31:0] = LDS offset
  unsigned long long gsrc = (unsigned long long)(uintptr_t)src;
  for (int off = (int)threadIdx.x * 16; off < nbytes; off += (int)blockDim.x * 16) {
    unsigned laddr = lds_base + (unsigned)off;
    asm volatile("global_load_async_to_lds_b128 %0, %1, %2"
                 :: "v"(laddr), "v"(off), "s"(gsrc)
                 : "memory");
  }
  asm volatile("s_wait_asynccnt 0" ::: "memory");
}

// Branchless bilinear sample, zero outside [0,255]^2 (map_coordinates order=1,
// cval=0): clamp tap addresses so loads are unconditional, mask with selects.
__device__ __forceinline__ float bilin(const float* img, float gx, float gy) {
  float fx = floorf(gx), fy = floorf(gy);
  float wx = gx - fx, wy = gy - fy;
  int ix = (int)fx, iy = (int)fy;
  int ix0 = min(max(ix, 0), NUMPIX - 1);
  int ix1 = min(max(ix + 1, 0), NUMPIX - 1);
  int iy0 = min(max(iy, 0), NUMPIX - 1);
  int iy1 = min(max(iy + 1, 0), NUMPIX - 1);
  const float* r0 = img + iy0 * NUMPIX;
  const float* r1 = img + iy1 * NUMPIX;
  float v00 = r0[ix0], v01 = r0[ix1];
  float v10 = r1[ix0], v11 = r1[ix1];
  bool bx0 = (unsigned)ix       < (unsigned)NUMPIX;
  bool bx1 = (unsigned)(ix + 1) < (unsigned)NUMPIX;
  bool by0 = (unsigned)iy       < (unsigned)NUMPIX;
  bool by1 = (unsigned)(iy + 1) < (unsigned)NUMPIX;
  v00 = (bx0 && by0) ? v00 : 0.f;
  v01 = (bx1 && by0) ? v01 : 0.f;
  v10 = (bx0 && by1) ? v10 : 0.f;
  v11 = (bx1 && by1) ? v11 : 0.f;
  float top = v00 + wx * (v01 - v00);
  float bot = v10 + wx * (v11 - v10);
  return top + wy * (bot - top);
}

// Forward projection of one ordered subset. One wave computes 16 detector bins
// for one angle; the 367-sample ray sum accumulates through
// V_WMMA_F32_16X16X4_F32 against an all-ones B (row-sum as matmul).
// Image (256 KB) lives in LDS for the whole kernel.
__global__ void fwd_kernel(const float* __restrict__ img,
                           const float* __restrict__ trig,
                           const float* __restrict__ sino,
                           const float* __restrict__ minv,
                           float* __restrict__ out,
                           float* __restrict__ res2,
                           int j, int mode) {
  extern __shared__ float ldsImg[];
  async_stage(img, ldsImg, IMG_N * 4);
  __syncthreads();

  int wave = threadIdx.x >> 5;
  int lane = threadIdx.x & 31;
  int tile = blockIdx.x * 8 + wave;            // 8 wave32 per 256-thread block
  if (tile >= NANG * STILES) return;
  int k  = tile / STILES;                      // local angle index in subset
  int st = tile % STILES;                      // 16-bin tile index
  int a  = j + NS * k;                         // global angle
  float ca = trig[a];
  float sa = trig[NUMTHETA + a];

  // WMMA f32 16x16x4 A layout: lanes 0-15 hold K=0,1; lanes 16-31 hold K=2,3.
  int m = lane & 15;                           // detector row M within tile
  int thalf = lane >> 4;                       // selects K pair {0,1} or {2,3}
  float sval = (float)(st * 16 + m) - CDET;
  float xs = sval * ca + CPIX;
  float ys = sval * sa + CPIX;

  v2f bmat = {1.0f, 1.0f};                     // all-ones B (4x16) => row sums
  v8f acc = {};
  float t0 = (float)(thalf * 2) - CDET;        // advances by 4 per chunk
  for (int chunk = 0; chunk < TCHUNKS; ++chunk, t0 += 4.0f) {
    float t1 = t0 + 1.0f;
    v2f amat;
    amat[0] = bilin(ldsImg, xs - t0 * sa, ys + t0 * ca);
    amat[1] = bilin(ldsImg, xs - t1 * sa, ys + t1 * ca);
    acc = __builtin_amdgcn_wmma_f32_16x16x4_f32(
        /*neg_a=*/false, amat, /*neg_b=*/false, bmat,
        /*c_mod=*/(short)0, acc, /*reuse_a=*/false, /*reuse_b=*/false);
  }

  // D[M][N] is constant over N (row sum). Lane 0 holds rows 0-7 in acc[0..7],
  // lane 16 holds rows 8-15.
  if (m == 0) {
    int rowbase = st * 16 + thalf * 8;
    if (mode == F_RES) {
      float local = 0.f;
      for (int v = 0; v < 8; ++v) {
        int sIdx = rowbase + v;
        if (sIdx < NUMBIN) {
          float d = acc[v] - sino[a * NUMBIN + sIdx];   // DX == 1
          local += d * d;
        }
      }
      atomicAdd(res2, local);
    } else if (mode == F_DIFFS) {
      for (int v = 0; v < 8; ++v) {
        int sIdx = rowbase + v;
        if (sIdx < NUMBIN)
          out[k * NUMBIN + sIdx] =
              (sino[a * NUMBIN + sIdx] - acc[v]) / minv[k * NUMBIN + sIdx];
      }
    } else { // F_MINV
      for (int v = 0; v < 8; ++v) {
        int sIdx = rowbase + v;
        if (sIdx < NUMBIN)
          out[k * NUMBIN + sIdx] = fmaxf(acc[v], DEN_EPS);
      }
    }
  }
}

// Backprojection of one subset sinogram; subset sinogram (~132 KB, padded to a
// 16B multiple for the async copy) + per-subset trig live in LDS.
__global__ void bp_kernel(const float* __restrict__ g,
                          const float* __restrict__ trig,
                          float* __restrict__ dinv_or_out,
                          float* __restrict__ f,
                          int j, int mode) {
  extern __shared__ float lds[];
  float* gl  = lds;              // FP_PAD sinogram values
  float* ct  = lds + FP_PAD;     // NANG cosines
  float* stg = ct + NANG;        // NANG sines
  async_stage(g, gl, FP_PAD * 4);          // <=16B over-read stays inside ws
  for (int i = threadIdx.x; i < NANG; i += blockDim.x) {
    int a = j + NS * i;
    ct[i]  = trig[a];
    stg[i] = trig[NUMTHETA + a];
  }
  __syncthreads();

  int p = blockIdx.x * blockDim.x + threadIdx.x;
  if (p >= IMG_N) return;
  float x = (float)(p & (NUMPIX - 1)) - CPIX;
  float y = (float)(p >> 8) - CPIX;
  float accv = 0.f;
  for (int k = 0; k < NANG; ++k) {
    float sd = x * ct[k] + y * stg[k] + CDET;
    float fi = floorf(sd);
    float w  = sd - fi;
    int i0 = (int)fi;
    int i0c = min(max(i0, 0), NUMBIN - 1);
    int i1c = min(max(i0 + 1, 0), NUMBIN - 1);
    const float* row = gl + k * NUMBIN;
    float v0 = row[i0c];
    float v1 = row[i1c];
    v0 = ((unsigned)i0       < (unsigned)NUMBIN) ? v0 : 0.f;
    v1 = ((unsigned)(i0 + 1) < (unsigned)NUMBIN) ? v1 : 0.f;
    accv += v0 + w * (v1 - v0);
  }
  if (mode == B_DINV) {
    dinv_or_out[p] = fmaxf(accv, DEN_EPS);
  } else { // B_SART: f += BETA * clamp(bp) / Dinv, BETA == 1
    float bp = (fabsf(accv) > 1000.f) ? 0.f : accv;
    f[p] += bp / dinv_or_out[p];
  }
}

__global__ void init_kernel(float* __restrict__ f, const float* __restrict__ f0,
                            float* __restrict__ ones_img,
                            float* __restrict__ ones_sino,
                            float* __restrict__ trig, float* __restrict__ res2) {
  int i = blockIdx.x * blockDim.x + threadIdx.x;
  if (i < IMG_N) { f[i] = f0[i]; ones_img[i] = 1.0f; }
  if (i < FP_N) ones_sino[i] = 1.0f;
  if (i < NUMTHETA) {
    float th = (float)((double)i * 0.017453292519943295); // deg -> rad
    trig[i] = cosf(th);
    trig[NUMTHETA + i] = sinf(th);
  }
  if (i == 0) res2[0] = 0.f;
}

__global__ void util_kernel(float* __restrict__ f, float* __restrict__ fprev,
                            float* __restrict__ res2, float* __restrict__ out,
                            int mode) {
  int i = blockIdx.x * blockDim.x + threadIdx.x;
  if (i >= IMG_N) return;
  if (mode == U_COPYPREV) {
    fprev[i] = f[i];
    if (i == 0) res2[0] = 0.f;
  } else if (mode == U_FMAXEPS) {
    f[i] = fmaxf(f[i], EPS64);
  } else if (mode == U_SELECT) {
    if (!(res2[0] > 1e-4f)) f[i] = fprev[i];   // res = sqrt(res2) <= 0.01 -> revert
  } else { // U_OUT
    out[i] = f[i];
  }
}

extern "C" void kernel_launch(void* const* d_in, const int* in_sizes, int n_in,
                              void* d_out, int out_size, void* d_ws, size_t ws_size,
                              hipStream_t stream) {
  const float* f0   = (const float*)d_in[0];   // [256,256] f32
  const float* sino = (const float*)d_in[1];   // [360,367] f32
  float* out = (float*)d_out;                  // [256,256] f32

  float* ws        = (float*)d_ws;             // ~2.6 MB of f32 scratch
  float* f         = ws;
  float* fprev     = f + IMG_N;
  float* ones_img  = fprev + IMG_N;
  float* diffs     = ones_img + IMG_N;
  float* ones_sino = diffs + FP_N;
  float* Minv      = ones_sino + FP_N;         // NS * FP_N
  float* Dinv      = Minv + NS * FP_N;         // NS * IMG_N
  float* trig      = Dinv + NS * IMG_N;        // 2 * NUMTHETA
  float* res2      = trig + 2 * NUMTHETA;      // 1

  const int    TB      = 256;                        // 8 wave32 per block
  const int    eGrid   = (IMG_N + TB - 1) / TB;      // 256
  const int    fwdGrid = (NANG * STILES + 7) / 8;    // 259
  const size_t fwdLds  = (size_t)IMG_N * sizeof(float);            // 256 KB image
  const size_t bpLds   = (size_t)(FP_PAD + 2 * NANG) * sizeof(float);

  init_kernel<<<eGrid, TB, 0, stream>>>(f, f0, ones_img, ones_sino, trig, res2);

  // Precompute per-subset normalizers (Minv = A_j(1), Dinv = A_j^T(1))
  for (int j = 0; j < NS; ++j)
    fwd_kernel<<<fwdGrid, TB, fwdLds, stream>>>(ones_img, trig, sino, nullptr,
                                                Minv + j * FP_N, res2, j, F_MINV);
  for (int j = 0; j < NS; ++j)
    bp_kernel<<<eGrid, TB, bpLds, stream>>>(ones_sino, trig, Dinv + j * IMG_N,
                                            nullptr, j, B_DINV);

  for (int it = 0; it < 2; ++it) {
    util_kernel<<<eGrid, TB, 0, stream>>>(f, fprev, res2, out, U_COPYPREV);
    for (int j = 0; j < NS; ++j) {
      fwd_kernel<<<fwdGrid, TB, fwdLds, stream>>>(f, trig, sino, Minv + j * FP_N,
                                                  diffs, res2, j, F_DIFFS);
      bp_kernel<<<eGrid, TB, bpLds, stream>>>(diffs, trig, Dinv + j * IMG_N,
                                              f, j, B_SART);
    }
    util_kernel<<<eGrid, TB, 0, stream>>>(f, fprev, res2, out, U_FMAXEPS);
    for (int j = 0; j < NS; ++j)
      fwd_kernel<<<fwdGrid, TB, fwdLds, stream>>>(f, trig, sino, nullptr,
                                                  nullptr, res2, j, F_RES);
    util_kernel<<<eGrid, TB, 0, stream>>>(f, fprev, res2, out, U_SELECT);
  }
  util_kernel<<<eGrid, TB, 0, stream>>>(f, fprev, res2, out, U_OUT);
}